// DiscreteGraphLearning_71914932404433
// MI455X (gfx1250) — compile-verified
//
#include <hip/hip_runtime.h>
#include <cstdint>

#define DEVFN static __device__ __forceinline__

// ---------------- problem dims ----------------
constexpr int B_ = 16, N_ = 207, T_ = 23990, P_ = 168, D_ = 96, EMB = 100, KNN = 10;
constexpr int L1 = T_ - 9;          // 23981 (conv1 output length)
constexpr int L2 = T_ - 18;         // 23972 (conv2 output length)
constexpr int DIM_FC = 16 * L2;     // 383552
constexpr int DIMH = P_ * D_;       // 16128
constexpr int NN = N_ * N_;         // 42849
constexpr int KN = KNN * N_;        // 2070
constexpr float BN_EPS = 1e-5f;

// ---------------- workspace layout (floats) ----------------
constexpr size_t OFF_CONV1 = 0;                                // [8][L1][N]
constexpr size_t SZ_CONV1  = (size_t)8 * L1 * N_;
constexpr size_t OFF_CONV2 = OFF_CONV1 + SZ_CONV1;             // [16][L2][N]
constexpr size_t SZ_CONV2  = (size_t)16 * L2 * N_;
constexpr size_t OFF_SMALL = OFF_CONV2 + SZ_CONV2;
// small region: [0..7]s1sum [8..15]s1sq [16..31]s2sum [32..47]s2sq
//               [48..55]sc1 [56..63]sh1 [64..79]sc2 [80..95]sh2
constexpr size_t OFF_XACC = OFF_SMALL + 128;                   // [N][EMB] fc accumulator
constexpr size_t OFF_XBN  = OFF_XACC + (size_t)N_ * EMB;       // [N][EMB] post-bn3
constexpr size_t OFF_PR   = OFF_XBN  + (size_t)N_ * EMB;       // [N][EMB]
constexpr size_t OFF_PS   = OFF_PR   + (size_t)N_ * EMB;       // [N][EMB]
constexpr size_t OFF_NORM = OFF_PS   + (size_t)N_ * EMB;       // [B][N]
constexpr size_t OFF_SIM  = OFF_NORM + (size_t)B_ * N_;        // [B][N][N]
constexpr size_t OFF_THR  = OFF_SIM  + (size_t)B_ * NN;        // [B]

// ---------------- output layout (floats, concatenated) ----------------
constexpr size_t O_BERN = 0;                                   // (B, N*N, 2)
constexpr size_t O_HID  = O_BERN + (size_t)B_ * NN * 2;        // (B, N, P, D)
constexpr size_t O_KNN  = O_HID  + (size_t)B_ * N_ * DIMH;     // (B, N, N)
constexpr size_t O_SADJ = O_KNN  + (size_t)B_ * NN;            // (B, N, N)

typedef float v2f __attribute__((ext_vector_type(2)));
typedef float v8f __attribute__((ext_vector_type(8)));

DEVFN int imin(int a, int b) { return a < b ? a : b; }

// ================= misc =================
__global__ void k_zero(float* w, size_t off, size_t cnt) {
  size_t i = (size_t)blockIdx.x * blockDim.x + threadIdx.x;
  if (i < cnt) w[off + i] = 0.0f;
}

// ================= conv1 + relu + bn1 stats =================
__global__ void k_conv1(const float* __restrict__ nf, const float* __restrict__ w1,
                        const float* __restrict__ b1, float* __restrict__ ws) {
  __shared__ float s_acc[16];
  int t = blockIdx.x, n = threadIdx.x;
  if (n < 16) s_acc[n] = 0.0f;
  __syncthreads();
  if (n < N_) {
    float x[10];
#pragma unroll
    for (int k = 0; k < 10; ++k) x[k] = nf[(size_t)(t + k) * N_ + n]; // coalesced over n
#pragma unroll
    for (int c = 0; c < 8; ++c) {
      float acc = b1[c];
#pragma unroll
      for (int k = 0; k < 10; ++k) acc += x[k] * w1[c * 10 + k];
      float y = fmaxf(acc, 0.0f);
      ws[OFF_CONV1 + ((size_t)c * L1 + t) * N_ + n] = y;
      atomicAdd(&s_acc[c], y);
      atomicAdd(&s_acc[8 + c], y * y);
    }
  }
  __syncthreads();
  if (n < 16) atomicAdd(&ws[OFF_SMALL + n], s_acc[n]);
}

// ================= BN finalize (stats -> affine fold) =================
__global__ void k_bnfin(float* ws, int nch, int sumoff, int outoff, float invcnt,
                        const float* __restrict__ g, const float* __restrict__ b) {
  int c = threadIdx.x;
  if (c < nch) {
    float m = ws[OFF_SMALL + sumoff + c] * invcnt;
    float v = ws[OFF_SMALL + sumoff + nch + c] * invcnt - m * m;
    float inv = rsqrtf(v + BN_EPS);
    ws[OFF_SMALL + outoff + c]       = g[c] * inv;            // scale
    ws[OFF_SMALL + outoff + nch + c] = b[c] - m * g[c] * inv; // shift
  }
}

// ================= conv2(bn1(x)) + relu + bn2 stats =================
__global__ void k_conv2(const float* __restrict__ w2, const float* __restrict__ b2,
                        float* __restrict__ ws) {
  __shared__ float sw[1280];
  __shared__ float s_acc[32];
  int t = blockIdx.x, n = threadIdx.x;
  for (int i = threadIdx.x; i < 1280; i += blockDim.x) sw[i] = w2[i];
  if (threadIdx.x < 32) s_acc[threadIdx.x] = 0.0f;
  __syncthreads();
  if (n < N_) {
    float sc[8], sh[8];
#pragma unroll
    for (int c = 0; c < 8; ++c) { sc[c] = ws[OFF_SMALL + 48 + c]; sh[c] = ws[OFF_SMALL + 56 + c]; }
    float a[8][10];
#pragma unroll
    for (int c = 0; c < 8; ++c)
#pragma unroll
      for (int k = 0; k < 10; ++k)
        a[c][k] = ws[OFF_CONV1 + ((size_t)c * L1 + t + k) * N_ + n] * sc[c] + sh[c];
#pragma unroll
    for (int c2 = 0; c2 < 16; ++c2) {
      float acc = b2[c2];
#pragma unroll
      for (int c = 0; c < 8; ++c)
#pragma unroll
        for (int k = 0; k < 10; ++k)
          acc += a[c][k] * sw[(c2 * 8 + c) * 10 + k];
      float y = fmaxf(acc, 0.0f);
      ws[OFF_CONV2 + ((size_t)c2 * L2 + t) * N_ + n] = y;
      atomicAdd(&s_acc[c2], y);
      atomicAdd(&s_acc[16 + c2], y * y);
    }
  }
  __syncthreads();
  if (threadIdx.x < 32) atomicAdd(&ws[OFF_SMALL + 16 + threadIdx.x], s_acc[threadIdx.x]);
}

// ====== fc GEMM via V_WMMA_F32_16X16X4_F32, K split + atomic reduce ======
// Branchless inner loop: invalid A rows / B cols only pollute C entries we never
// store, so indices are clamped instead of masked (no EXEC churn in the loop).
// BN channel select uses a hoisted boundary compare (chunk < L2 crosses <= 1).
constexpr int FC_KSPLIT = 64;
constexpr int FC_CHUNK  = 5996;   // multiple of 4; 64*5996 >= DIM_FC; DIM_FC%4==0

__global__ __launch_bounds__(32) void k_fc_wmma(const float* __restrict__ fcw,
                                                float* __restrict__ ws) {
  const int lane = threadIdx.x;
  const int hi   = lane >> 4;        // 0 | 1
  const int sub  = lane & 15;
  const int ksel = hi * 2;           // lanes 0-15 -> K0,K1 ; lanes 16-31 -> K2,K3
  const int n = imin(blockIdx.x * 16 + sub, N_ - 1);   // clamped A row (node)
  const int e = imin(blockIdx.y * 16 + sub, EMB - 1);  // clamped B col (embedding)
  const float* wrow = fcw + (size_t)e * DIM_FC;
  const int f0 = blockIdx.z * FC_CHUNK;
  const int f1 = imin(f0 + FC_CHUNK, DIM_FC);
  // BN2 fold: channel of feature f is f / L2; hoist to one boundary compare
  const int cbase = f0 / L2;
  const int cnext = imin(cbase + 1, 15);
  const int boundary = (cbase + 1) * L2;
  const float scLo = ws[OFF_SMALL + 64 + cbase], scHi = ws[OFF_SMALL + 64 + cnext];
  const float shLo = ws[OFF_SMALL + 80 + cbase], shHi = ws[OFF_SMALL + 80 + cnext];
  const float* abase = ws + OFF_CONV2 + n;   // + f*N_ per feature
  v8f acc = {0, 0, 0, 0, 0, 0, 0, 0};
  for (int f = f0; f < f1; f += 4) {
    const int fa0 = f + ksel;
    float r0 = abase[(size_t)fa0 * N_];
    float r1 = abase[(size_t)(fa0 + 1) * N_];
    float s0 = (fa0     >= boundary) ? scHi : scLo;
    float t0 = (fa0     >= boundary) ? shHi : shLo;
    float s1 = (fa0 + 1 >= boundary) ? scHi : scLo;
    float t1 = (fa0 + 1 >= boundary) ? shHi : shLo;
    v2f av = { r0 * s0 + t0, r1 * s1 + t1 };
    v2f bv = *(const v2f*)&wrow[fa0];                   // contiguous -> b64 load
    __builtin_prefetch(wrow + f + 512, 0, 1);           // global_prefetch_b8
    acc = __builtin_amdgcn_wmma_f32_16x16x4_f32(false, av, false, bv, (short)0, acc,
                                                false, false);
  }
#pragma unroll
  for (int r = 0; r < 8; ++r) {
    int m = blockIdx.x * 16 + r + hi * 8;   // C/D layout: vgpr r, lanes>=16 -> M+8
    int col = blockIdx.y * 16 + sub;
    if (m < N_ && col < EMB)
      atomicAdd(&ws[OFF_XACC + (size_t)m * EMB + col], acc[r]);
  }
}

// ================= relu(fc+bias) then bn3 over nodes =================
__global__ void k_xfin(const float* __restrict__ fcb, const float* __restrict__ g3,
                       const float* __restrict__ b3, float* __restrict__ ws) {
  __shared__ float ssum, ssq, smean, sinv;
  int e = blockIdx.x, n = threadIdx.x;
  if (n == 0) { ssum = 0.f; ssq = 0.f; }
  __syncthreads();
  float v = 0.f; bool act = n < N_;
  if (act) {
    v = fmaxf(ws[OFF_XACC + (size_t)n * EMB + e] + fcb[e], 0.0f);
    atomicAdd(&ssum, v); atomicAdd(&ssq, v * v);
  }
  __syncthreads();
  if (n == 0) {
    float m = ssum / (float)N_;
    float var = ssq / (float)N_ - m * m;
    smean = m; sinv = rsqrtf(var + BN_EPS);
  }
  __syncthreads();
  if (act) ws[OFF_XBN + (size_t)n * EMB + e] = (v - smean) * sinv * g3[e] + b3[e];
}

// ================= ps = x@W_s.T , pr = x@W_r.T =================
__global__ void k_prps(const float* __restrict__ fow, float* __restrict__ ws) {
  int idx = blockIdx.x * blockDim.x + threadIdx.x;
  if (idx >= N_ * EMB * 2) return;
  int which = idx / (N_ * EMB);            // 0 = ps (W_s), 1 = pr (W_r)
  int rem = idx - which * (N_ * EMB);
  int n = rem / EMB, e = rem - (rem / EMB) * EMB;
  const float* w = fow + (size_t)e * (2 * EMB) + (which ? EMB : 0);
  const float* x = ws + OFF_XBN + (size_t)n * EMB;
  float acc = 0.f;
  for (int j = 0; j < EMB; ++j) acc += x[j] * w[j];
  ws[(which ? OFF_PR : OFF_PS) + (size_t)n * EMB + e] = acc;
}

// ================= JAX threefry2x32, key(42) = (0, 42) =================
DEVFN uint32_t rotl32(uint32_t x, int r) { return (x << r) | (x >> (32 - r)); }
struct U2 { uint32_t a, b; };
DEVFN U2 threefry(uint32_t x0, uint32_t x1) {
  const uint32_t ks0 = 0u, ks1 = 42u, ks2 = 0x1BD11BDAu ^ 42u;
  x0 += ks0; x1 += ks1;
#define RND(r) { x0 += x1; x1 = rotl32(x1, r); x1 ^= x0; }
  RND(13) RND(15) RND(26) RND(6)   x0 += ks1; x1 += ks2 + 1u;
  RND(17) RND(29) RND(16) RND(24)  x0 += ks2; x1 += ks0 + 2u;
  RND(13) RND(15) RND(26) RND(6)   x0 += ks0; x1 += ks1 + 3u;
  RND(17) RND(29) RND(16) RND(24)  x0 += ks1; x1 += ks2 + 4u;
  RND(13) RND(15) RND(26) RND(6)   x0 += ks2; x1 += ks0 + 5u;
#undef RND
  return {x0, x1};
}
DEVFN float gumbel_at(uint32_t flat) {
  const uint32_t half = (uint32_t)((size_t)B_ * NN);  // 685584 = total/2
  uint32_t r = (flat < half) ? threefry(flat, flat + half).a
                             : threefry(flat - half, flat).b;
  float u = __uint_as_float((r >> 9) | 0x3f800000u) - 1.0f;
  return -logf(-logf(u + 1e-20f) + 1e-20f);
}

// ================= edge + logits + gumbel-softmax hard sample =================
__global__ void k_edge(const float* __restrict__ fob, const float* __restrict__ fcatw,
                       const float* __restrict__ fcatb, const float* __restrict__ ws,
                       float* __restrict__ out) {
  int idx = blockIdx.x * blockDim.x + threadIdx.x;
  if (idx >= NN) return;
  int i = idx / N_, j = idx - (idx / N_) * N_;
  const float* pri = ws + OFF_PR + (size_t)i * EMB;
  const float* psj = ws + OFF_PS + (size_t)j * EMB;
  float l0 = fcatb[0], l1 = fcatb[1];
  for (int e = 0; e < EMB; ++e) {
    float ed = fmaxf(pri[e] + psj[e] + fob[e], 0.0f);
    l0 += ed * fcatw[e];
    l1 += ed * fcatw[EMB + e];
  }
  for (int b = 0; b < B_; ++b) {
    size_t pos = (size_t)b * NN + idx;
    out[O_BERN + pos * 2 + 0] = l0;
    out[O_BERN + pos * 2 + 1] = l1;
    float g0 = gumbel_at((uint32_t)(pos * 2 + 0));
    float g1 = gumbel_at((uint32_t)(pos * 2 + 1));
    float z0 = (l0 + g0) * 2.0f, z1 = (l1 + g1) * 2.0f;   // / TEMP (0.5)
    float mx = fmaxf(z0, z1);
    float e0 = expf(z0 - mx), e1 = expf(z1 - mx);
    float s0 = e0 / (e0 + e1), s1 = e1 / (e0 + e1);
    float h0 = (s0 >= s1) ? 1.0f : 0.0f;                  // argmax, first on tie
    float y0 = (h0 - s0) + s0;                            // straight-through fwd
    out[O_SADJ + pos] = (i == j) ? 0.0f : y0;
  }
}

// ================= row norms of hidden =================
__global__ void k_norm(const float* __restrict__ h, float* __restrict__ ws) {
  __shared__ float s;
  if (threadIdx.x == 0) s = 0.f;
  __syncthreads();
  const float* p = h + (size_t)blockIdx.x * DIMH;
  float acc = 0.f;
  for (int k = threadIdx.x; k < DIMH; k += blockDim.x) { float v = p[k]; acc += v * v; }
  atomicAdd(&s, acc);
  __syncthreads();
  if (threadIdx.x == 0) ws[OFF_NORM + blockIdx.x] = sqrtf(s) + 1e-10f;
}

// ====== sim = (h @ h.T)/(norm_i*norm_j): async-LDS staged WMMA f32 ======
// A/B k-panels (16 x 64 f32 each) are staged with GLOBAL_LOAD_ASYNC_TO_LDS_B128
// (ASYNCcnt) then consumed by v_wmma_f32_16x16x4_f32 from LDS (ds_load_b64).
constexpr int KP = 64;   // k-panel width (floats); DIMH % KP == 0

__global__ __launch_bounds__(32) void k_sim_wmma(const float* __restrict__ h,
                                                 float* __restrict__ ws) {
  __shared__ float smA[16 * KP];
  __shared__ float smB[16 * KP];
  const int lane = threadIdx.x;
  const int hi = lane >> 4, sub = lane & 15, ksel = hi * 2;
  const int b = blockIdx.z;
  const float* hb = h + (size_t)b * N_ * DIMH;
  // staging: instruction q copies rows {q*2+hi}, cols sub*4..sub*4+3 (16B/lane)
  const float* aPtr[8]; const float* bPtr[8];
  uint32_t aLds[8], bLds[8];
#pragma unroll
  for (int q = 0; q < 8; ++q) {
    int ra = imin(blockIdx.x * 16 + q * 2 + hi, N_ - 1);
    int rb = imin(blockIdx.y * 16 + q * 2 + hi, N_ - 1);
    aPtr[q] = hb + (size_t)ra * DIMH + sub * 4;
    bPtr[q] = hb + (size_t)rb * DIMH + sub * 4;
    aLds[q] = (uint32_t)(uintptr_t)&smA[(q * 2 + hi) * KP + sub * 4];
    bLds[q] = (uint32_t)(uintptr_t)&smB[(q * 2 + hi) * KP + sub * 4];
  }
  v8f acc = {0, 0, 0, 0, 0, 0, 0, 0};
  for (int kp = 0; kp < DIMH; kp += KP) {
#pragma unroll
    for (int q = 0; q < 8; ++q) {
      asm volatile("global_load_async_to_lds_b128 %0, %1, off"
                   :: "v"(aLds[q]), "v"(aPtr[q] + kp) : "memory");
      asm volatile("global_load_async_to_lds_b128 %0, %1, off"
                   :: "v"(bLds[q]), "v"(bPtr[q] + kp) : "memory");
    }
    asm volatile("s_wait_asynccnt 0x0" ::: "memory");
#pragma unroll
    for (int kk = 0; kk < KP; kk += 4) {
      v2f av = *(const v2f*)&smA[sub * KP + kk + ksel];   // ds_load_b64
      v2f bv = *(const v2f*)&smB[sub * KP + kk + ksel];
      acc = __builtin_amdgcn_wmma_f32_16x16x4_f32(false, av, false, bv, (short)0, acc,
                                                  false, false);
    }
  }
#pragma unroll
  for (int r = 0; r < 8; ++r) {
    int mi = blockIdx.x * 16 + r + hi * 8;
    int cj = blockIdx.y * 16 + sub;
    if (mi < N_ && cj < N_) {
      float inv = 1.0f / (ws[OFF_NORM + (size_t)b * N_ + mi] *
                          ws[OFF_NORM + (size_t)b * N_ + cj]);
      ws[OFF_SIM + (size_t)b * NN + (size_t)mi * N_ + cj] = acc[r] * inv;
    }
  }
}

// ================= per-batch top-(K*N) threshold via histogram ================
__global__ void k_topk_thresh(float* __restrict__ ws) {
  __shared__ int hist[2048];
  int b = blockIdx.x;
  for (int i = threadIdx.x; i < 2048; i += blockDim.x) hist[i] = 0;
  __syncthreads();
  const float* sim = ws + OFF_SIM + (size_t)b * NN;
  for (int idx = threadIdx.x; idx < NN; idx += blockDim.x) {
    float v = sim[idx];
    int bin = (int)floorf((v + 1.0f) * 1024.0f);
    bin = bin < 0 ? 0 : (bin > 2047 ? 2047 : bin);
    atomicAdd(&hist[bin], 1);
  }
  __syncthreads();
  if (threadIdx.x == 0) {
    int cum = 0, bt = 0;
    for (int i = 2047; i >= 0; --i) { cum += hist[i]; if (cum >= KN) { bt = i; break; } }
    ws[OFF_THR + b] = (float)bt * (1.0f / 1024.0f) - 1.0f;
  }
}

// ================= adj_knn =================
__global__ void k_adjknn(const float* __restrict__ ws, float* __restrict__ out) {
  size_t idx = (size_t)blockIdx.x * blockDim.x + threadIdx.x;
  if (idx >= (size_t)B_ * NN) return;
  int b = (int)(idx / NN);
  int rem = (int)(idx - (size_t)b * NN);
  int i = rem / N_, j = rem - (rem / N_) * N_;
  float v = ws[OFF_SIM + idx];
  float thr = ws[OFF_THR + b];
  out[O_KNN + idx] = (i != j && v != 0.0f && v >= thr) ? 1.0f : 0.0f;
}

// ================= launch =================
extern "C" void kernel_launch(void* const* d_in, const int* in_sizes, int n_in,
                              void* d_out, int out_size, void* d_ws, size_t ws_size,
                              hipStream_t stream) {
  (void)in_sizes; (void)n_in; (void)out_size; (void)ws_size;
  const float* hid   = (const float*)d_in[0];
  const float* nf    = (const float*)d_in[1];
  const float* c1w   = (const float*)d_in[2];
  const float* c1b   = (const float*)d_in[3];
  const float* c2w   = (const float*)d_in[4];
  const float* c2b   = (const float*)d_in[5];
  const float* bn1g  = (const float*)d_in[6];
  const float* bn1b  = (const float*)d_in[7];
  const float* bn2g  = (const float*)d_in[8];
  const float* bn2b  = (const float*)d_in[9];
  const float* fcw   = (const float*)d_in[10];
  const float* fcb   = (const float*)d_in[11];
  const float* bn3g  = (const float*)d_in[12];
  const float* bn3b  = (const float*)d_in[13];
  // d_in[14], d_in[15] (fc_mean_*) feed `_dyn`, which is dead in the reference.
  const float* fow   = (const float*)d_in[16];
  const float* fob   = (const float*)d_in[17];
  const float* fcatw = (const float*)d_in[18];
  const float* fcatb = (const float*)d_in[19];
  float* ws  = (float*)d_ws;
  float* out = (float*)d_out;

  { // zero stats + fc accumulator (re-done every launch: graph-replay safe)
    size_t cnt = 128 + (size_t)N_ * EMB;
    k_zero<<<(unsigned)((cnt + 255) / 256), 256, 0, stream>>>(ws, OFF_SMALL, cnt);
  }
  k_conv1<<<L1, 256, 0, stream>>>(nf, c1w, c1b, ws);
  k_bnfin<<<1, 32, 0, stream>>>(ws, 8, 0, 48, 1.0f / (float)((size_t)N_ * L1), bn1g, bn1b);
  k_conv2<<<L2, 256, 0, stream>>>(c2w, c2b, ws);
  k_bnfin<<<1, 32, 0, stream>>>(ws, 16, 16, 64, 1.0f / (float)((size_t)N_ * L2), bn2g, bn2b);
  k_fc_wmma<<<dim3(13, 7, FC_KSPLIT), 32, 0, stream>>>(fcw, ws);
  k_xfin<<<EMB, 256, 0, stream>>>(fcb, bn3g, bn3b, ws);
  k_prps<<<(N_ * EMB * 2 + 255) / 256, 256, 0, stream>>>(fow, ws);
  k_edge<<<(NN + 255) / 256, 256, 0, stream>>>(fob, fcatw, fcatb, ws, out);
  hipMemcpyAsync(out + O_HID, hid, (size_t)B_ * N_ * DIMH * sizeof(float),
                 hipMemcpyDeviceToDevice, stream);
  k_norm<<<B_ * N_, 256, 0, stream>>>(hid, ws);
  k_sim_wmma<<<dim3(13, 13, B_), 32, 0, stream>>>(hid, ws);
  k_topk_thresh<<<B_, 1024, 0, stream>>>(ws);
  k_adjknn<<<(unsigned)(((size_t)B_ * NN + 255) / 256), 256, 0, stream>>>(ws, out);
}